// user_encoder_76501957476778
// MI455X (gfx1250) — compile-verified
//
#include <hip/hip_runtime.h>
#include <math.h>

// ---------------------------------------------------------------------------
// GNN user-encoder forward for MI455X (gfx1250, wave32).
// All dense contractions use V_WMMA_F32_16X16X4_F32 (exact fp32 WMMA).
// GEMM is compile-time specialized (transA / bias / act) with a branch-free
// unrolled fast path for interior tiles + global_prefetch_b8 streaming hints.
// Attention softmax / column softmax / gathers are small VALU kernels.
// ---------------------------------------------------------------------------

typedef __attribute__((ext_vector_type(2))) float v2f;
typedef __attribute__((ext_vector_type(8))) float v8f;

#define HEADS 20
#define MAXO  20

// ---------------------------------------------------------------------------
// C = act(op(A) * B + bias), one wave (32 lanes) per 16x16 output tile.
// K consumed 4 per WMMA per the CDNA5 16x4 (A) / 4x16 (B) fragment layouts:
//   A frag: lane 0-15 = rows, half-wave selects K pair {0,1} vs {2,3}
//   B frag: lane = column, same half-wave K-pair split
//   C/D   : VGPR i -> row (i + 8*half), lane = column
// TRANSA: 0 A[M,K], 1 A stored [K,M] (lda = row stride of storage)
// BIASMODE: 0 none, 1 bias[col], 2 bias[row];  ACT: 0 none, 1 tanh
// grid.z batches heads via element strides sA/sB/sC.
// ---------------------------------------------------------------------------
template <int TRANSA, int BIASMODE, int ACT>
__global__ __launch_bounds__(32)
void wmma_gemm_t(const float* __restrict__ Ag, const float* __restrict__ Bg,
                 const float* __restrict__ bias, float* __restrict__ Cg,
                 int M, int N, int K, int lda, int ldb, int ldc,
                 long long sA, long long sB, long long sC)
{
    const float* Ap = Ag + (size_t)blockIdx.z * sA;
    const float* Bp = Bg + (size_t)blockIdx.z * sB;
    float*       Cp = Cg + (size_t)blockIdx.z * sC;

    const int tileN = blockIdx.x * 16;
    const int tileM = blockIdx.y * 16;
    const int lane  = threadIdx.x;        // 0..31
    const int half  = lane >> 4;          // K-pair select
    const int lid   = lane & 15;

    const int rowA = tileM + lid;         // A fragment row (M)
    const int colB = tileN + lid;         // B fragment col (N)

    v8f acc = {};

    const bool fullTile = (tileM + 16 <= M) && (tileN + 16 <= N) && ((K & 3) == 0);

    if (fullTile) {
        // -------- branch-free interior path --------
        const float* arow = TRANSA ? (Ap + rowA)                  // stride lda per K
                                   : (Ap + (size_t)rowA * lda);   // stride 1  per K
        const float* bcol = Bp + colB;                            // stride ldb per K
#pragma unroll 4
        for (int k0 = 0; k0 < K; k0 += 4) {
            const int kk = k0 + 2 * half;
            v2f a, b;
            if (TRANSA) {
                a.x = arow[(size_t)kk * lda];
                a.y = arow[(size_t)(kk + 1) * lda];
            } else {
                a.x = arow[kk];
                a.y = arow[kk + 1];
            }
            b.x = bcol[(size_t)kk * ldb];
            b.y = bcol[(size_t)(kk + 1) * ldb];
            // stream hint for the strided B walk (speculative: dropped if OOB)
            __builtin_prefetch(bcol + (size_t)(kk + 16) * ldb, 0, 1);
            acc = __builtin_amdgcn_wmma_f32_16x16x4_f32(
                      false, a, false, b, (short)0, acc, false, false);
        }
#pragma unroll
        for (int i = 0; i < 8; i++) {
            const int cm = tileM + i + 8 * half;
            const int cn = tileN + lid;
            float v = acc[i];
            if (BIASMODE == 1)      v += bias[cn];
            else if (BIASMODE == 2) v += bias[cm];
            if (ACT == 1) v = tanhf(v);
            Cp[(size_t)cm * ldc + cn] = v;
        }
    } else {
        // -------- guarded edge path --------
        for (int k0 = 0; k0 < K; k0 += 4) {
            const int kk = k0 + 2 * half;
            v2f a = {0.f, 0.f};
            v2f b = {0.f, 0.f};
            if (rowA < M) {
                if (kk < K)
                    a.x = TRANSA ? Ap[(size_t)kk * lda + rowA]
                                 : Ap[(size_t)rowA * lda + kk];
                if (kk + 1 < K)
                    a.y = TRANSA ? Ap[(size_t)(kk + 1) * lda + rowA]
                                 : Ap[(size_t)rowA * lda + (kk + 1)];
            }
            if (colB < N) {
                if (kk < K)     b.x = Bp[(size_t)kk * ldb + colB];
                if (kk + 1 < K) b.y = Bp[(size_t)(kk + 1) * ldb + colB];
            }
            acc = __builtin_amdgcn_wmma_f32_16x16x4_f32(
                      false, a, false, b, (short)0, acc, false, false);
        }
        const int cn = tileN + lid;
        if (cn < N) {
#pragma unroll
            for (int i = 0; i < 8; i++) {
                const int cm = tileM + i + 8 * half;
                if (cm < M) {
                    float v = acc[i];
                    if (BIASMODE == 1)      v += bias[cn];
                    else if (BIASMODE == 2) v += bias[cm];
                    if (ACT == 1) v = tanhf(v);
                    Cp[(size_t)cm * ldc + cn] = v;
                }
            }
        }
    }
}

// ---------------------------------------------------------------------------
__global__ void zero_kernel(float* __restrict__ p, int n)
{
    int i = blockIdx.x * blockDim.x + threadIdx.x;
    if (i < n) p[i] = 0.f;
}

__global__ void gather_rows_kernel(const float* __restrict__ emb,
                                   const int* __restrict__ idx,
                                   float* __restrict__ out, int rows, int dim)
{
    int i = blockIdx.x * blockDim.x + threadIdx.x;
    if (i >= rows * dim) return;
    int r = i / dim, c = i % dim;
    out[i] = emb[(size_t)idx[r] * dim + c];
}

// f1[h,i] = h[h,i,:].a1[h,:]   f2[h,i] = h[h,i,:].a2[h,:]
__global__ void gat_f_kernel(const float* __restrict__ h,
                             const float* __restrict__ a1,
                             const float* __restrict__ a2,
                             float* __restrict__ f1, float* __restrict__ f2,
                             int n, int o, int H)
{
    int idx = blockIdx.x * blockDim.x + threadIdx.x;
    if (idx >= H * n) return;
    int hd = idx / n, i = idx % n;
    const float* hv  = h  + ((size_t)hd * n + i) * o;
    const float* a1v = a1 + (size_t)hd * o;
    const float* a2v = a2 + (size_t)hd * o;
    float s1 = 0.f, s2 = 0.f;
    for (int k = 0; k < o; k++) { s1 += hv[k] * a1v[k]; s2 += hv[k] * a2v[k]; }
    f1[idx] = s1;
    f2[idx] = s2;
}

// One block per (row, head): masked leaky-relu softmax over neighbors and
// weighted aggregation of h. Deterministic fixed-order block reductions.
__global__ __launch_bounds__(128)
void gat_attn_kernel(const float* __restrict__ h,   // [H,n,o]
                     const float* __restrict__ f1,  // [H,n]
                     const float* __restrict__ f2,  // [H,n]
                     const float* __restrict__ Aadj,// [n,n]
                     float* __restrict__ out,       // [H,n,o]
                     int n, int o)
{
    const int row = blockIdx.x;
    const int hd  = blockIdx.y;
    const int tid = threadIdx.x;
    const int BLK = blockDim.x;

    const float* hh   = h + (size_t)hd * n * o;
    const float* Arow = Aadj + (size_t)row * n;
    const float* f2h  = f2 + (size_t)hd * n;
    const float  f1v  = f1[(size_t)hd * n + row];

    __shared__ float red[128];
    __shared__ float s_mx, s_sum;
    __shared__ float part[128 * MAXO];

    // pass 1: row max of masked leaky_relu(f1+f2)
    float mx = -1e30f;
    for (int m = tid; m < n; m += BLK) {
        float e;
        if (Arow[m] > 0.f) { float t = f1v + f2h[m]; e = (t > 0.f) ? t : 0.2f * t; }
        else               e = -1e9f;
        mx = fmaxf(mx, e);
    }
    red[tid] = mx; __syncthreads();
    for (int s = BLK >> 1; s > 0; s >>= 1) {
        if (tid < s) red[tid] = fmaxf(red[tid], red[tid + s]);
        __syncthreads();
    }
    if (tid == 0) s_mx = red[0];
    __syncthreads();
    mx = s_mx;

    // pass 2: exp-sum + weighted accumulation of h rows
    float acc[MAXO];
    for (int i = 0; i < MAXO; i++) acc[i] = 0.f;
    float ssum = 0.f;
    for (int m = tid; m < n; m += BLK) {
        float e;
        if (Arow[m] > 0.f) { float t = f1v + f2h[m]; e = (t > 0.f) ? t : 0.2f * t; }
        else               e = -1e9f;
        float w = expf(e - mx);
        ssum += w;
        const float* hm = hh + (size_t)m * o;
        for (int i = 0; i < o; i++) acc[i] += w * hm[i];
    }
    red[tid] = ssum; __syncthreads();
    for (int s = BLK >> 1; s > 0; s >>= 1) {
        if (tid < s) red[tid] += red[tid + s];
        __syncthreads();
    }
    if (tid == 0) s_sum = red[0];
    __syncthreads();
    const float inv = 1.f / s_sum;

    for (int i = 0; i < o; i++) part[tid * o + i] = acc[i];
    __syncthreads();
    if (tid < o) {
        float s = 0.f;
        for (int t = 0; t < BLK; t++) s += part[t * o + tid];
        out[((size_t)hd * n + row) * o + tid] = s * inv;
    }
}

// mean over heads (concat=0) or per-head concat (concat=1)
__global__ void gat_combine_kernel(const float* __restrict__ tmp,  // [H,n,o]
                                   float* __restrict__ out,
                                   int n, int o, int H, int concat, int total)
{
    int idx = blockIdx.x * blockDim.x + threadIdx.x;
    if (idx >= total) return;
    if (concat) {
        int k = idx % o; int t = idx / o;
        int h = t % H;   int i = t / H;
        out[(size_t)i * (H * o) + h * o + k] = tmp[((size_t)h * n + i) * o + k];
    } else {
        int k = idx % o; int i = idx / o;
        float s = 0.f;
        for (int h = 0; h < H; h++) s += tmp[((size_t)h * n + i) * o + k];
        out[idx] = s / (float)H;
    }
}

// Column-wise softmax over m rows of src[m,k]; write (with row padding offset)
// into dst[(row_off+r)*ldd + col_off + col]. One block per column.
__global__ __launch_bounds__(128)
void col_softmax_pad_kernel(const float* __restrict__ src, int m, int k,
                            float* __restrict__ dst, int ldd,
                            int row_off, int col_off)
{
    const int col = blockIdx.x;
    const int tid = threadIdx.x;
    const int BLK = blockDim.x;
    __shared__ float red[128];
    __shared__ float s_mx, s_sum;

    float mx = -1e30f;
    for (int r = tid; r < m; r += BLK) mx = fmaxf(mx, src[(size_t)r * k + col]);
    red[tid] = mx; __syncthreads();
    for (int s = BLK >> 1; s > 0; s >>= 1) {
        if (tid < s) red[tid] = fmaxf(red[tid], red[tid + s]);
        __syncthreads();
    }
    if (tid == 0) s_mx = red[0];
    __syncthreads();
    mx = s_mx;

    float sum = 0.f;
    for (int r = tid; r < m; r += BLK) sum += expf(src[(size_t)r * k + col] - mx);
    red[tid] = sum; __syncthreads();
    for (int s = BLK >> 1; s > 0; s >>= 1) {
        if (tid < s) red[tid] += red[tid + s];
        __syncthreads();
    }
    if (tid == 0) s_sum = red[0];
    __syncthreads();
    const float inv = 1.f / s_sum;

    for (int r = tid; r < m; r += BLK)
        dst[(size_t)(row_off + r) * ldd + col_off + col] =
            expf(src[(size_t)r * k + col] - mx) * inv;
}

// ---------------------------------------------------------------------------
// Host-side GEMM dispatch (compile-time specialization selection)
// ---------------------------------------------------------------------------
static void launch_gemm(const float* A, const float* B, const float* bias, float* C,
                        int M, int N, int K, int lda, int ldb, int ldc,
                        long long sA, long long sB, long long sC,
                        int transA, int biasMode, int act, int batch,
                        hipStream_t stream)
{
    dim3 grid((N + 15) / 16, (M + 15) / 16, batch);
#define GEMM_CASE(T, BM, AC)                                                    \
    wmma_gemm_t<T, BM, AC><<<grid, 32, 0, stream>>>(A, B, bias, C, M, N, K,     \
                                                    lda, ldb, ldc, sA, sB, sC)
    if (!transA) {
        if (biasMode == 1 && act == 1)      GEMM_CASE(0, 1, 1);
        else if (biasMode == 1)             GEMM_CASE(0, 1, 0);
        else                                GEMM_CASE(0, 0, 0);
    } else {
        if (biasMode == 2)                  GEMM_CASE(1, 2, 0);
        else                                GEMM_CASE(1, 0, 0);
    }
#undef GEMM_CASE
}

// ---------------------------------------------------------------------------
// Workspace layout (float offsets)
// ---------------------------------------------------------------------------
static const size_t WS_X     = 0;        // x [1100,400]
static const size_t WS_HTMP  = 440064;   // h  [20,1100,20]
static const size_t WS_ATMP  = 880128;   // attn out [20,1100,20]
static const size_t WS_F1    = 1320192;  // [20,1100]
static const size_t WS_F2    = 1342208;  // [20,1100]
static const size_t WS_BASIS = 1364224;  // [1100,20]
static const size_t WS_GBUF  = 1386240;  // gat outputs, <=1100*16
static const size_t WS_PMAT  = 1403904;  // assign pmat, <=1000*16
static const size_t WS_P1    = 1419968;  // [1100,28]
static const size_t WS_APB   = 1450816;  // A*P [1100,28]
static const size_t WS_AP1   = 1481664;  // [28,28]
static const size_t WS_XP1   = 1482496;  // [28,20]
static const size_t WS_P2    = 1483072;  // [28,3]
static const size_t WS_APB2  = 1483200;  // Ap1*P2 [28,3]
static const size_t WS_AP2   = 1483328;  // [3,3]
static const size_t WS_XP2   = 1483392;  // [3,20]
static const size_t WS_DP    = 1483456;  // [3,20]
static const size_t WS_THID  = 1483520;  // [1000,100]
static const size_t WS_TGAT  = 1583520;  // [1000,100]
// total ~1,683,520 floats (~6.8 MB)

extern "C" void kernel_launch(void* const* d_in, const int* in_sizes, int n_in,
                              void* d_out, int out_size, void* d_ws, size_t ws_size,
                              hipStream_t stream)
{
    (void)in_sizes; (void)n_in; (void)out_size; (void)ws_size;
    float* ws = (float*)d_ws;

    const int NN = 1100, NNEWS = 50, NCAT = 1000, NENT = 50;

    // ---- inputs (setup_inputs order; params dict flattened in insertion order)
    const float* title   = (const float*)d_in[0];   // [50,768]
    const float* Aadj    = (const float*)d_in[6];   // [1,1100,1100]
    const int*   catidx  = (const int*)  d_in[7];   // [1000]
    const float* entemb  = (const float*)d_in[8];   // [50,100]
    const float* Wt      = (const float*)d_in[9];   // [768,400]
    const float* bt      = (const float*)d_in[10];  // [400]
    const float* cat_emb = (const float*)d_in[11];  // [1000,100]
    const float* Wc1     = (const float*)d_in[12];  const float* bc1 = (const float*)d_in[13];
    const float* Wc2     = (const float*)d_in[14];  const float* bc2 = (const float*)d_in[15];
    const float* We1     = (const float*)d_in[16];  const float* be1 = (const float*)d_in[17];
    const float* We2     = (const float*)d_in[18];  const float* be2 = (const float*)d_in[19];
    // gat param triples: W [H,Fin,o], a1 [H,o], a2 [H,o]
    // 0:basis(o=20) 1:n1(16) 2:t1(3) 3:e1(9) 4:n2(1) 5:t2(1) 6:e2(1) 7:dp(1)
    const float* gW[8];  const float* ga1[8]; const float* ga2[8];
    for (int g = 0; g < 8; g++) {
        gW[g]  = (const float*)d_in[20 + 3 * g + 0];
        ga1[g] = (const float*)d_in[20 + 3 * g + 1];
        ga2[g] = (const float*)d_in[20 + 3 * g + 2];
    }
    const float* fc_n1_W = (const float*)d_in[44]; const float* fc_n1_b = (const float*)d_in[45];
    const float* fc_t1_W = (const float*)d_in[46]; const float* fc_t1_b = (const float*)d_in[47];
    const float* fc_e1_W = (const float*)d_in[48]; const float* fc_e1_b = (const float*)d_in[49];
    const float* fc_n2_W = (const float*)d_in[50]; const float* fc_n2_b = (const float*)d_in[51];
    const float* fc_t2_W = (const float*)d_in[52]; const float* fc_t2_b = (const float*)d_in[53];
    const float* fc_e2_W = (const float*)d_in[54]; const float* fc_e2_b = (const float*)d_in[55];

    auto gemm = [&](const float* A, const float* B, const float* bias, float* C,
                    int M, int N, int K, int lda, int ldb, int ldc,
                    long long sA, long long sB, long long sC,
                    int transA, int biasMode, int act, int batch) {
        launch_gemm(A, B, bias, C, M, N, K, lda, ldb, ldc, sA, sB, sC,
                    transA, biasMode, act, batch, stream);
    };

    auto run_gat = [&](const float* x, int n, int Fin, const float* Am,
                       int g, int o, float* out, int concat) {
        // h[h] = x @ W[h]   (batched over heads)
        gemm(x, gW[g], nullptr, ws + WS_HTMP, n, o, Fin, Fin, o, o,
             0, (long long)Fin * o, (long long)n * o, 0, 0, 0, HEADS);
        int tot = HEADS * n;
        gat_f_kernel<<<(tot + 255) / 256, 256, 0, stream>>>(
            ws + WS_HTMP, ga1[g], ga2[g], ws + WS_F1, ws + WS_F2, n, o, HEADS);
        dim3 ga(n, HEADS);
        gat_attn_kernel<<<ga, 128, 0, stream>>>(
            ws + WS_HTMP, ws + WS_F1, ws + WS_F2, Am, ws + WS_ATMP, n, o);
        int ct = concat ? n * HEADS * o : n * o;
        gat_combine_kernel<<<(ct + 255) / 256, 256, 0, stream>>>(
            ws + WS_ATMP, out, n, o, HEADS, concat, ct);
    };

    auto run_assign = [&](const float* g /*[n,k]*/, int n, int k,
                          const float* W /*[n,m]*/, const float* b, int m,
                          int lo, float* Pdst, int ldP, int coloff) {
        // pmat[mi,kc] = sum_n W[n,mi] g[n,kc] + b[mi]  -> W^T @ g (transA)
        gemm(W, g, b, ws + WS_PMAT, m, k, n, m, k, k, 0, 0, 0, 1, 2, 0, 1);
        col_softmax_pad_kernel<<<k, 128, 0, stream>>>(
            ws + WS_PMAT, m, k, Pdst, ldP, lo, coloff);
    };

    auto run_pool = [&](const float* Am, int n, const float* P, int Kc,
                        const float* xin, int F, float* APb, float* Apo, float* xpo) {
        gemm(Am, P,  nullptr, APb, n,  Kc, n, n,  Kc, Kc, 0, 0, 0, 0, 0, 0, 1); // AP = A@P
        gemm(P, APb, nullptr, Apo, Kc, Kc, n, Kc, Kc, Kc, 0, 0, 0, 1, 0, 0, 1); // Ap = P^T AP
        gemm(P, xin, nullptr, xpo, Kc, F,  n, Kc, F,  F,  0, 0, 0, 1, 0, 0, 1); // xp = P^T x
    };

    // ---- zero padded assignment buffers
    zero_kernel<<<(NN * 28 + 255) / 256, 256, 0, stream>>>(ws + WS_P1, NN * 28);
    zero_kernel<<<(28 * 3 + 255) / 256, 256, 0, stream>>>(ws + WS_P2, 28 * 3);

    // ---- node feature encoders -> x [1100,400]
    gemm(title, Wt, bt, ws + WS_X, NNEWS, 400, 768, 768, 400, 400, 0,0,0, 0, 1, 1, 1);
    gather_rows_kernel<<<(NCAT * 100 + 255) / 256, 256, 0, stream>>>(
        cat_emb, catidx, ws + WS_TGAT, NCAT, 100);
    gemm(ws + WS_TGAT, Wc1, bc1, ws + WS_THID, NCAT, 100, 100, 100, 100, 100, 0,0,0, 0, 1, 0, 1);
    gemm(ws + WS_THID, Wc2, bc2, ws + WS_X + (size_t)NNEWS * 400, NCAT, 400, 100,
         100, 400, 400, 0,0,0, 0, 1, 1, 1);
    gemm(entemb, We1, be1, ws + WS_GBUF, NENT, 100, 100, 100, 100, 100, 0,0,0, 0, 1, 0, 1);
    gemm(ws + WS_GBUF, We2, be2, ws + WS_X + (size_t)(NNEWS + NCAT) * 400, NENT, 400, 100,
         100, 400, 400, 0,0,0, 0, 1, 1, 1);

    // ---- level-1 GATs + assignments + pooling
    run_gat(ws + WS_X, NN, 400, Aadj, 0, 20, ws + WS_BASIS, 0);

    run_gat(ws + WS_BASIS, NN, 20, Aadj, 1, 16, ws + WS_GBUF, 0);
    run_assign(ws + WS_GBUF, NN, 16, fc_n1_W, fc_n1_b, 50,   0,    ws + WS_P1, 28, 0);
    run_gat(ws + WS_BASIS, NN, 20, Aadj, 2, 3, ws + WS_GBUF, 0);
    run_assign(ws + WS_GBUF, NN, 3,  fc_t1_W, fc_t1_b, 1000, 50,   ws + WS_P1, 28, 16);
    run_gat(ws + WS_BASIS, NN, 20, Aadj, 3, 9, ws + WS_GBUF, 0);
    run_assign(ws + WS_GBUF, NN, 9,  fc_e1_W, fc_e1_b, 50,   1050, ws + WS_P1, 28, 19);

    run_pool(Aadj, NN, ws + WS_P1, 28, ws + WS_BASIS, 20,
             ws + WS_APB, ws + WS_AP1, ws + WS_XP1);

    // ---- level-2 GATs + assignments + pooling  (n=28, adjacency = Ap1)
    run_gat(ws + WS_XP1, 28, 20, ws + WS_AP1, 4, 1, ws + WS_GBUF, 0);
    run_assign(ws + WS_GBUF, 28, 1, fc_n2_W, fc_n2_b, 16, 0,  ws + WS_P2, 3, 0);
    run_gat(ws + WS_XP1, 28, 20, ws + WS_AP1, 5, 1, ws + WS_GBUF, 0);
    run_assign(ws + WS_GBUF, 28, 1, fc_t2_W, fc_t2_b, 3,  16, ws + WS_P2, 3, 1);
    run_gat(ws + WS_XP1, 28, 20, ws + WS_AP1, 6, 1, ws + WS_GBUF, 0);
    run_assign(ws + WS_GBUF, 28, 1, fc_e2_W, fc_e2_b, 9,  19, ws + WS_P2, 3, 2);

    run_pool(ws + WS_AP1, 28, ws + WS_P2, 3, ws + WS_XP1, 20,
             ws + WS_APB2, ws + WS_AP2, ws + WS_XP2);

    // ---- final head-concat GAT + softmax + readout
    run_gat(ws + WS_XP2, 3, 20, ws + WS_AP2, 7, 1, ws + WS_GBUF, 1); // [3,20] concat
    col_softmax_pad_kernel<<<20, 128, 0, stream>>>(ws + WS_GBUF, 3, 20,
                                                   ws + WS_DP, 20, 0, 0);
    // user[k,f] = sum_n dp[n,k] * xp2[n,f]  -> dp^T @ xp2 -> d_out [20,20]
    gemm(ws + WS_DP, ws + WS_XP2, nullptr, (float*)d_out,
         20, 20, 3, 20, 20, 20, 0, 0, 0, 1, 0, 0, 1);
}